// BondMessagePassing_6021544149649
// MI455X (gfx1250) — compile-verified
//
#include <hip/hip_runtime.h>

#define N_NODES_C 50000
#define N_EDGES_C 800000
#define NODE_F_C 128
#define BOND_F_C 16
#define HID_C    128
#define KIN_C    (NODE_F_C + BOND_F_C)   // 144
#define KOUT_C   (NODE_F_C + HID_C)      // 256

// Padded LDS strides (floats): row starts 16B-aligned, bank-conflict-free
// for the 16-lane B-fragment reads (stride mod 64 = 20 / 4 / 4).
#define LDK_IN 148   // for K=144
#define LDK_H  132   // for K=128
#define LDK_O  260   // for K=256

typedef float v2f __attribute__((ext_vector_type(2)));
typedef float v8f __attribute__((ext_vector_type(8)));

__device__ __forceinline__ v2f ld2(const float* __restrict__ p) {
  return *(const v2f* __restrict__)p;
}

__device__ __forceinline__ v8f wmma_f32(v2f a, v2f b, v8f c) {
  // D = A(16x4 f32) * B(4x16 f32) + C(16x16 f32)
  return __builtin_amdgcn_wmma_f32_16x16x4_f32(
      /*neg_a=*/false, a, /*neg_b=*/false, b,
      /*c_mod=*/(short)0, c, /*reuse_a=*/false, /*reuse_b=*/false);
}

// ---------------------------------------------------------------------------
// Stage a [HID x K] row-major weight matrix into LDS (stride LDL floats)
// using CDNA5 async global->LDS B128 copies (ASYNCcnt), then sync.
// All 256 threads of the block must participate.
// ---------------------------------------------------------------------------
template <int K, int LDL>
__device__ __forceinline__ void stage_weights(const float* __restrict__ g,
                                              float* __restrict__ l) {
  constexpr int chunksPerRow = K / 4;           // 16B chunks
  constexpr int total = HID_C * chunksPerRow;   // exact multiple of 256
  for (int i = threadIdx.x; i < total; i += 256) {
    const int n = i / chunksPerRow;
    const int c = i - n * chunksPerRow;
    const float* gs = g + (size_t)n * K + (c << 2);
    float* ld = l + (size_t)n * LDL + (c << 2);
    const unsigned lo = (unsigned)(uintptr_t)ld;           // LDS byte offset
    const unsigned long long ga = (unsigned long long)(uintptr_t)gs;
    asm volatile("global_load_async_to_lds_b128 %0, %1, off"
                 :: "v"(lo), "v"(ga) : "memory");
  }
  asm volatile("s_wait_asynccnt 0" ::: "memory");
  __syncthreads();
}

// ---------------------------------------------------------------------------
// Kernel 1: H0 = concat(x[src], edge_attr) @ W_i^T + b_i ; Hc = relu(H0)
// One wave computes a 16-edge x 128-channel tile.
// Two-stage ping-pong pipeline (trip count 36 is even) -> no rotation moves.
// ---------------------------------------------------------------------------
__global__ __launch_bounds__(256) void gemm_in_kernel(
    const float* __restrict__ x, const float* __restrict__ ea,
    const int* __restrict__ src, const float* __restrict__ Wi,
    const float* __restrict__ bi, float* __restrict__ H0,
    float* __restrict__ Hc)
{
  extern __shared__ float Wlds[];
  stage_weights<KIN_C, LDK_IN>(Wi, Wlds);

  const int wave  = threadIdx.x >> 5;
  const int lane  = threadIdx.x & 31;
  const int tile  = blockIdx.x * 8 + wave;        // 50000 tiles exactly
  const int ebase = tile * 16;
  const int row   = lane & 15;                    // A row / B column-in-tile
  const int khalf = (lane >> 4) << 1;             // 0 or 2
  const int e     = ebase + row;

  const float* __restrict__ xrow = x  + (size_t)src[e] * NODE_F_C;
  const float* __restrict__ erow = ea + (size_t)e      * BOND_F_C;

  const v8f vzero = {0.f,0.f,0.f,0.f,0.f,0.f,0.f,0.f};
  v8f acc[8];
  #pragma unroll
  for (int i = 0; i < 8; ++i) acc[i] = vzero;

  auto loadA = [&](int kk) -> v2f {
    return (kk < NODE_F_C) ? ld2(xrow + kk) : ld2(erow + (kk - NODE_F_C));
  };
  auto loadB = [&](int kk, v2f* b) {
    #pragma unroll
    for (int nt = 0; nt < 8; ++nt)
      b[nt] = ld2(&Wlds[(nt * 16 + row) * LDK_IN + kk]);
  };
  auto mma8 = [&](v2f a, const v2f* b) {
    #pragma unroll
    for (int nt = 0; nt < 8; ++nt) acc[nt] = wmma_f32(a, b[nt], acc[nt]);
  };

  constexpr int STEPS = KIN_C / 4;                // 36 (even)
  v2f a0 = loadA(khalf), a1;
  v2f b0[8], b1[8];
  loadB(khalf, b0);

  #pragma unroll 1
  for (int ks = 0; ks < STEPS - 2; ks += 2) {
    const int k1 = (ks + 1) * 4 + khalf;
    const int k2 = (ks + 2) * 4 + khalf;
    a1 = loadA(k1); loadB(k1, b1);
    mma8(a0, b0);
    a0 = loadA(k2); loadB(k2, b0);
    mma8(a1, b1);
  }
  {
    const int k1 = (STEPS - 1) * 4 + khalf;
    a1 = loadA(k1); loadB(k1, b1);
    mma8(a0, b0);
    mma8(a1, b1);
  }

  const int mhi = (lane >> 4) << 3;               // 0 or 8
  #pragma unroll
  for (int nt = 0; nt < 8; ++nt) {
    const int n = nt * 16 + row;
    const float bias = bi[n];
    #pragma unroll
    for (int rr = 0; rr < 8; ++rr) {
      const int m = rr + mhi;
      const size_t idx = (size_t)(ebase + m) * HID_C + n;
      const float pre = acc[nt][rr] + bias;
      H0[idx] = pre;
      Hc[idx] = pre > 0.f ? pre : 0.f;
    }
  }
}

// ---------------------------------------------------------------------------
// Zero the node aggregation buffer (N_NODES * HID floats).
// ---------------------------------------------------------------------------
__global__ __launch_bounds__(256) void zero_kernel(float* __restrict__ p) {
  p[(size_t)blockIdx.x * 256 + threadIdx.x] = 0.f;
}

// ---------------------------------------------------------------------------
// Scatter-sum: agg[dst[e], :] += H[e, :]. 4 channels per thread.
// ---------------------------------------------------------------------------
__global__ __launch_bounds__(256) void scatter_kernel(
    const float* __restrict__ H, const int* __restrict__ dst,
    float* __restrict__ agg)
{
  const size_t tid = (size_t)blockIdx.x * 256 + threadIdx.x;  // E*32 threads
  const int e  = (int)(tid >> 5);
  const int c0 = ((int)tid & 31) << 2;
  float* __restrict__ arow = agg + (size_t)dst[e] * HID_C + c0;
  const float* __restrict__ hrow = H + (size_t)e * HID_C + c0;
  #pragma unroll
  for (int j = 0; j < 4; ++j) atomicAdd(arow + j, hrow[j]);
}

// ---------------------------------------------------------------------------
// Kernel 3: M = agg[src] - Hin[rev];  Hout = relu(H0 + M @ W_h^T + b_h)
// ---------------------------------------------------------------------------
__global__ __launch_bounds__(256) void edge_update_kernel(
    const float* __restrict__ agg, const int* __restrict__ src,
    const int* __restrict__ rev, const float* __restrict__ H0,
    const float* __restrict__ Wh, const float* __restrict__ bh,
    const float* __restrict__ Hin, float* __restrict__ Hout)
{
  extern __shared__ float Wlds[];
  stage_weights<HID_C, LDK_H>(Wh, Wlds);

  const int wave  = threadIdx.x >> 5;
  const int lane  = threadIdx.x & 31;
  const int tile  = blockIdx.x * 8 + wave;
  const int ebase = tile * 16;
  const int row   = lane & 15;
  const int khalf = (lane >> 4) << 1;
  const int e     = ebase + row;

  const float* __restrict__ arow = agg + (size_t)src[e] * HID_C;
  const float* __restrict__ hrow = Hin + (size_t)rev[e] * HID_C;

  const v8f vzero = {0.f,0.f,0.f,0.f,0.f,0.f,0.f,0.f};
  v8f acc[8];
  #pragma unroll
  for (int i = 0; i < 8; ++i) acc[i] = vzero;

  auto loadA = [&](int kk) -> v2f {
    return ld2(arow + kk) - ld2(hrow + kk);       // M = agg[src] - Hin[rev]
  };
  auto loadB = [&](int kk, v2f* b) {
    #pragma unroll
    for (int nt = 0; nt < 8; ++nt)
      b[nt] = ld2(&Wlds[(nt * 16 + row) * LDK_H + kk]);
  };
  auto mma8 = [&](v2f a, const v2f* b) {
    #pragma unroll
    for (int nt = 0; nt < 8; ++nt) acc[nt] = wmma_f32(a, b[nt], acc[nt]);
  };

  constexpr int STEPS = HID_C / 4;                // 32 (even)
  v2f a0 = loadA(khalf), a1;
  v2f b0[8], b1[8];
  loadB(khalf, b0);

  #pragma unroll 1
  for (int ks = 0; ks < STEPS - 2; ks += 2) {
    const int k1 = (ks + 1) * 4 + khalf;
    const int k2 = (ks + 2) * 4 + khalf;
    a1 = loadA(k1); loadB(k1, b1);
    mma8(a0, b0);
    a0 = loadA(k2); loadB(k2, b0);
    mma8(a1, b1);
  }
  {
    const int k1 = (STEPS - 1) * 4 + khalf;
    a1 = loadA(k1); loadB(k1, b1);
    mma8(a0, b0);
    mma8(a1, b1);
  }

  const int mhi = (lane >> 4) << 3;
  #pragma unroll
  for (int nt = 0; nt < 8; ++nt) {
    const int n = nt * 16 + row;
    const float bias = bh[n];
    #pragma unroll
    for (int rr = 0; rr < 8; ++rr) {
      const int m = rr + mhi;
      const size_t idx = (size_t)(ebase + m) * HID_C + n;
      const float v = acc[nt][rr] + bias + H0[idx];
      Hout[idx] = v > 0.f ? v : 0.f;
    }
  }
}

// ---------------------------------------------------------------------------
// Kernel 4: Mn[n,:] = (rowsum(agg[n,:]) == 0) ? x[n,:] : agg[n,:]
// One wave per node; wave-level reduction over 128 channels.
// ---------------------------------------------------------------------------
__global__ __launch_bounds__(256) void node_mask_kernel(
    const float* __restrict__ agg, const float* __restrict__ x,
    float* __restrict__ Mn)
{
  const int wave = threadIdx.x >> 5;
  const int lane = threadIdx.x & 31;
  const int node = blockIdx.x * 8 + wave;         // 50000 nodes exactly
  const size_t base = (size_t)node * HID_C + (size_t)lane * 4;

  const float4 v = *(const float4*)(agg + base);
  float s = v.x + v.y + v.z + v.w;
  #pragma unroll
  for (int off = 16; off > 0; off >>= 1) s += __shfl_xor(s, off, 32);

  float4 o;
  if (s == 0.f) o = *(const float4*)(x + base);   // NODE_F == HID
  else          o = v;
  *(float4*)(Mn + base) = o;
}

// ---------------------------------------------------------------------------
// Kernel 5: out = relu(concat(x, Mn) @ W_o^T + b_o)
// ---------------------------------------------------------------------------
__global__ __launch_bounds__(256) void node_out_kernel(
    const float* __restrict__ x, const float* __restrict__ Mn,
    const float* __restrict__ Wo, const float* __restrict__ bo,
    float* __restrict__ out)
{
  extern __shared__ float Wlds[];
  stage_weights<KOUT_C, LDK_O>(Wo, Wlds);   // all threads, before the guard

  const int wave = threadIdx.x >> 5;
  const int lane = threadIdx.x & 31;
  const int tile = blockIdx.x * 8 + wave;
  if (tile >= N_NODES_C / 16) return;             // uniform per wave
  const int nbase = tile * 16;
  const int row   = lane & 15;
  const int khalf = (lane >> 4) << 1;

  const float* __restrict__ xrow = x  + (size_t)(nbase + row) * NODE_F_C;
  const float* __restrict__ mrow = Mn + (size_t)(nbase + row) * HID_C;

  const v8f vzero = {0.f,0.f,0.f,0.f,0.f,0.f,0.f,0.f};
  v8f acc[8];
  #pragma unroll
  for (int i = 0; i < 8; ++i) acc[i] = vzero;

  auto loadA = [&](int kk) -> v2f {
    return (kk < NODE_F_C) ? ld2(xrow + kk) : ld2(mrow + (kk - NODE_F_C));
  };
  auto loadB = [&](int kk, v2f* b) {
    #pragma unroll
    for (int nt = 0; nt < 8; ++nt)
      b[nt] = ld2(&Wlds[(nt * 16 + row) * LDK_O + kk]);
  };
  auto mma8 = [&](v2f a, const v2f* b) {
    #pragma unroll
    for (int nt = 0; nt < 8; ++nt) acc[nt] = wmma_f32(a, b[nt], acc[nt]);
  };

  constexpr int STEPS = KOUT_C / 4;               // 64 (even)
  v2f a0 = loadA(khalf), a1;
  v2f b0[8], b1[8];
  loadB(khalf, b0);

  #pragma unroll 1
  for (int ks = 0; ks < STEPS - 2; ks += 2) {
    const int k1 = (ks + 1) * 4 + khalf;
    const int k2 = (ks + 2) * 4 + khalf;
    a1 = loadA(k1); loadB(k1, b1);
    mma8(a0, b0);
    a0 = loadA(k2); loadB(k2, b0);
    mma8(a1, b1);
  }
  {
    const int k1 = (STEPS - 1) * 4 + khalf;
    a1 = loadA(k1); loadB(k1, b1);
    mma8(a0, b0);
    mma8(a1, b1);
  }

  const int mhi = (lane >> 4) << 3;
  #pragma unroll
  for (int nt = 0; nt < 8; ++nt) {
    const int n = nt * 16 + row;
    const float bias = bo[n];
    #pragma unroll
    for (int rr = 0; rr < 8; ++rr) {
      const int m = rr + mhi;
      const float v = acc[nt][rr] + bias;
      out[(size_t)(nbase + m) * HID_C + n] = v > 0.f ? v : 0.f;
    }
  }
}

// ---------------------------------------------------------------------------
extern "C" void kernel_launch(void* const* d_in, const int* in_sizes, int n_in,
                              void* d_out, int out_size, void* d_ws, size_t ws_size,
                              hipStream_t stream) {
  (void)in_sizes; (void)n_in; (void)out_size; (void)ws_size;
  const float* x   = (const float*)d_in[0];
  const float* ea  = (const float*)d_in[1];
  const int*   ei  = (const int*)d_in[2];    // [2, E] row-major: src then dst
  const int*   rev = (const int*)d_in[3];
  const float* Wi  = (const float*)d_in[4];
  const float* bi  = (const float*)d_in[5];
  const float* Wh  = (const float*)d_in[6];
  const float* bh  = (const float*)d_in[7];
  const float* Wo  = (const float*)d_in[8];
  const float* bo  = (const float*)d_in[9];
  float* out = (float*)d_out;

  const int* src = ei;
  const int* dst = ei + N_EDGES_C;

  // Workspace layout (floats): H0 | Ha | Hb | agg | Mn
  float* H0  = (float*)d_ws;
  float* Ha  = H0  + (size_t)N_EDGES_C * HID_C;
  float* Hb  = Ha  + (size_t)N_EDGES_C * HID_C;
  float* agg = Hb  + (size_t)N_EDGES_C * HID_C;
  float* Mn  = agg + (size_t)N_NODES_C * HID_C;

  const int edgeTileBlocks = (N_EDGES_C / 16) / 8;              // 6250
  const int zeroBlocks     = (N_NODES_C * HID_C) / 256;         // 25000
  const int scatterBlocks  = (N_EDGES_C * (HID_C / 4)) / 256;   // 100000
  const int nodeMaskBlocks = N_NODES_C / 8;                     // 6250
  const int nodeOutBlocks  = (N_NODES_C / 16 + 7) / 8;          // 391

  const size_t smemIn  = (size_t)HID_C * LDK_IN * sizeof(float); //  75776 B
  const size_t smemH   = (size_t)HID_C * LDK_H  * sizeof(float); //  67584 B
  const size_t smemO   = (size_t)HID_C * LDK_O  * sizeof(float); // 133120 B

  gemm_in_kernel<<<edgeTileBlocks, 256, smemIn, stream>>>(x, ea, src, Wi, bi, H0, Ha);

  float* Hin = Ha; float* Hout = Hb;
  for (int it = 0; it < 2; ++it) {                // DEPTH-1 message passes
    zero_kernel<<<zeroBlocks, 256, 0, stream>>>(agg);
    scatter_kernel<<<scatterBlocks, 256, 0, stream>>>(Hin, dst, agg);
    edge_update_kernel<<<edgeTileBlocks, 256, smemH, stream>>>(
        agg, src, rev, H0, Wh, bh, Hin, Hout);
    float* t = Hin; Hin = Hout; Hout = t;
  }

  zero_kernel<<<zeroBlocks, 256, 0, stream>>>(agg);
  scatter_kernel<<<scatterBlocks, 256, 0, stream>>>(Hin, dst, agg);
  node_mask_kernel<<<nodeMaskBlocks, 256, 0, stream>>>(agg, x, Mn);
  node_out_kernel<<<nodeOutBlocks, 256, smemO, stream>>>(x, Mn, Wo, bo, out);
}